// MarketImpactLayer_50397146251289
// MI455X (gfx1250) — compile-verified
//
#include <hip/hip_runtime.h>

// MI455X / gfx1250, wave32. Full-f32 pipeline:
//  - scalar market-impact math (VALU)
//  - GEMM1 [16x3]x[3x64]+ReLU via per-lane FMAs (K=3, WMMA not worth it)
//  - GEMM2 [16x64]x[64x32] via chained V_WMMA_F32_16X16X4_F32 (f32-precise)
//  - LayerNorm(32) via intra-half shfl_xor reductions
// Memory floor ~296 MB @ 23.3 TB/s ~= 12.7 us; WMMA keeps GEMM2 off the
// scalar VALU pipe so we stay near that bound.

typedef __attribute__((ext_vector_type(2))) float v2f;
typedef __attribute__((ext_vector_type(8))) float v8f;

#define GAMMA_C 0.1f
#define ETA_C   0.3f

__global__ __launch_bounds__(256) void market_impact_wmma_kernel(
    const float* __restrict__ os_in,  const float* __restrict__ liq_in,
    const float* __restrict__ W1,     const float* __restrict__ b1,
    const float* __restrict__ W2,     const float* __restrict__ b2,
    const float* __restrict__ lnw,    const float* __restrict__ lnb,
    float* __restrict__ out, int nrows)
{
    __shared__ float sW1[3 * 64];     // row-major [3][64]
    __shared__ float sB1[64];
    __shared__ float sW2[64 * 32];    // row-major [64][32]
    __shared__ float sB2[32];
    __shared__ float sLnW[32];
    __shared__ float sLnB[32];

    const int tid = threadIdx.x;
    for (int i = tid; i < 192;  i += 256) sW1[i] = W1[i];
    for (int i = tid; i < 64;   i += 256) sB1[i] = b1[i];
    for (int i = tid; i < 2048; i += 256) sW2[i] = W2[i];
    if (tid < 32) { sB2[tid] = b2[tid]; sLnW[tid] = lnw[tid]; sLnB[tid] = lnb[tid]; }
    __syncthreads();

    const int lane = tid & 31;
    const int wave = tid >> 5;
    const int r    = lane & 15;    // row within 16-row tile
    const int sel  = lane >> 4;    // lane-half: selects K pair / row group
    const int n0   = r;            // output column within an N-tile

    const int tile = blockIdx.x * 8 + wave;
    const int row0 = tile * 16;
    if (row0 >= nrows) return;     // wave-uniform guard (B divides evenly anyway)
    const int gr = row0 + r;

    // ---- per-row impact features (both lane-halves compute the same row r) ----
    const float os  = os_in[gr];
    const float liq = liq_in[gr];
    const float perm  = GAMMA_C * os / liq;
    const float temp  = ETA_C * copysignf(sqrtf(fabsf(os)), os) / sqrtf(liq);
    const float total = perm + temp;

    // ---- GEMM1 (K=3) + bias + ReLU, producing A fragments for WMMA ----
    // f32 16x4 A layout: for K-step j, VGPR0 = A[r][4j+2*sel], VGPR1 = A[r][4j+2*sel+1]
    v2f a[16];
#pragma unroll
    for (int j = 0; j < 16; ++j) {
        const int c0 = 4 * j + 2 * sel;
        float h0 = sB1[c0]     + perm * sW1[c0]     + temp * sW1[64 + c0]     + total * sW1[128 + c0];
        float h1 = sB1[c0 + 1] + perm * sW1[c0 + 1] + temp * sW1[64 + c0 + 1] + total * sW1[128 + c0 + 1];
        a[j].x = fmaxf(h0, 0.0f);
        a[j].y = fmaxf(h1, 0.0f);
    }

    // ---- B fragments from W2 (4x16 per K-step, two N-tiles) ----
    v2f bf0[16], bf1[16];
#pragma unroll
    for (int j = 0; j < 16; ++j) {
        const int k0 = 4 * j + 2 * sel;
        bf0[j].x = sW2[k0 * 32 + n0];
        bf0[j].y = sW2[(k0 + 1) * 32 + n0];
        bf1[j].x = sW2[k0 * 32 + 16 + n0];
        bf1[j].y = sW2[(k0 + 1) * 32 + 16 + n0];
    }

    // ---- accumulators seeded with bias b2 (column-only) ----
    v8f c0v, c1v;
    const float bias0 = sB2[n0], bias1 = sB2[16 + n0];
#pragma unroll
    for (int i = 0; i < 8; ++i) { c0v[i] = bias0; c1v[i] = bias1; }

    // ---- GEMM2: K=64 as 16 chained f32 WMMAs per N-tile (EXEC all-1s here) ----
#pragma unroll
    for (int j = 0; j < 16; ++j)
        c0v = __builtin_amdgcn_wmma_f32_16x16x4_f32(false, a[j], false, bf0[j],
                                                    (short)0, c0v, false, false);
#pragma unroll
    for (int j = 0; j < 16; ++j)
        c1v = __builtin_amdgcn_wmma_f32_16x16x4_f32(false, a[j], false, bf1[j],
                                                    (short)0, c1v, false, false);

    // ---- LayerNorm over 32 cols per row; row(i) = i + 8*sel, cols split n0 / 16+n0 ----
    const float lw0 = sLnW[n0], lw1 = sLnW[16 + n0];
    const float lb0 = sLnB[n0], lb1 = sLnB[16 + n0];
    float* __restrict__ enc = out + (size_t)nrows * 3;

#pragma unroll
    for (int i = 0; i < 8; ++i) {
        float x0 = c0v[i], x1 = c1v[i];
        float s  = x0 + x1;
        float sq = x0 * x0 + x1 * x1;
        // reduce across the 16-lane half (xor masks 1,2,4,8 never cross bit 4)
#pragma unroll
        for (int m = 1; m <= 8; m <<= 1) {
            s  += __shfl_xor(s,  m, 32);
            sq += __shfl_xor(sq, m, 32);
        }
        const float mean = s * 0.03125f;                       // /32
        const float var  = fmaxf(sq * 0.03125f - mean * mean, 0.0f);
        const float inv  = rsqrtf(var + 1e-5f);
        const float y0 = (x0 - mean) * inv * lw0 + lb0;
        const float y1 = (x1 - mean) * inv * lw1 + lb1;
        const int grow = row0 + i + 8 * sel;
        enc[(size_t)grow * 32 + n0]      = y0;
        enc[(size_t)grow * 32 + 16 + n0] = y1;
    }

    // ---- scalar outputs (divergent, after all WMMAs) ----
    if (sel == 0) {
        out[gr]                       = perm;
        out[(size_t)nrows + gr]       = temp;
        out[(size_t)2 * nrows + gr]   = total;
    }
}

extern "C" void kernel_launch(void* const* d_in, const int* in_sizes, int n_in,
                              void* d_out, int out_size, void* d_ws, size_t ws_size,
                              hipStream_t stream) {
    const float* os  = (const float*)d_in[0];
    const float* liq = (const float*)d_in[1];
    const float* W1  = (const float*)d_in[2];
    const float* b1  = (const float*)d_in[3];
    const float* W2  = (const float*)d_in[4];
    const float* b2  = (const float*)d_in[5];
    const float* lnw = (const float*)d_in[6];
    const float* lnb = (const float*)d_in[7];
    float* out = (float*)d_out;

    const int nrows = in_sizes[0];               // B = 2,000,000
    const int tiles = (nrows + 15) / 16;         // 16 rows per wave
    const int blocks = (tiles + 7) / 8;          // 8 waves (256 threads) per block

    market_impact_wmma_kernel<<<blocks, 256, 0, stream>>>(
        os, liq, W1, b1, W2, b2, lnw, lnb, out, nrows);
}